// SoftArgMax2d_4956392259702
// MI455X (gfx1250) — compile-verified
//
#include <hip/hip_runtime.h>
#include <hip/hip_bf16.h>
#include <stdint.h>

// Spatial soft-argmax: x[B,C,128,128] fp32 -> out[B,C,2] fp32
//   m  = max_hw(x);  e = exp(x - m)
//   ex = sum(e * col/127) / sum(e);  ey = sum(e * row/127) / sum(e)
//
// Strategy (MI455X / gfx1250):
//  * 1 workgroup per (b,c) heatmap: 64 KB tile -> LDS via async copy
//    (GLOBAL_LOAD_ASYNC_TO_LDS_B128, ASYNCcnt), single HBM read.
//  * pass 1: block max from LDS (wave32 shfl_xor + LDS scratch)
//  * pass 2: exp via v_exp_f32, accumulate {sum, sum*col, sum*row}
//  * bandwidth-bound (128 MiB @ 23.3 TB/s ~ 5.5 us) -> WMMA not applicable.

#define MAP_ELEMS   16384   // 128*128
#define MAP_W       128
#define NTHREADS    256     // 8 wave32
#define NWAVES      8
#define F4_PER_THR  16      // 16384 floats / 4 / 256 threads

__global__ __launch_bounds__(NTHREADS)
void soft_argmax2d_kernel(const float* __restrict__ x, float* __restrict__ out,
                          int nmaps) {
    extern __shared__ float tile[];     // MAP_ELEMS floats (64 KB, dynamic)
    __shared__ float red[32];           // [0..7] max, [8..15] s, [16..23] sx, [24..31] sy

    const int tid = threadIdx.x;
    const int bc  = blockIdx.x;
    if (bc >= nmaps) return;

    const float* __restrict__ src = x + (size_t)bc * MAP_ELEMS;

    // ---------------- async copy: global -> LDS (b128 per lane) -------------
#pragma unroll
    for (int i = 0; i < F4_PER_THR; ++i) {
        const int f4 = i * NTHREADS + tid;                       // float4 index
        const unsigned long long gaddr =
            (unsigned long long)(uintptr_t)(src + f4 * 4);
        const unsigned lds_addr =
            (unsigned)(uintptr_t)(tile + f4 * 4);                // low 32 bits = LDS offset
        asm volatile("global_load_async_to_lds_b128 %0, %1, off"
                     :: "v"(lds_addr), "v"(gaddr)
                     : "memory");
    }
    asm volatile("s_wait_asynccnt 0x0" ::: "memory");
    __syncthreads();

    const float4* __restrict__ t4 = reinterpret_cast<const float4*>(tile);

    // ---------------- pass 1: block max --------------------------------------
    float m = -3.402823466e+38f;
#pragma unroll
    for (int i = 0; i < F4_PER_THR; ++i) {
        const float4 v = t4[i * NTHREADS + tid];
        m = fmaxf(m, fmaxf(fmaxf(v.x, v.y), fmaxf(v.z, v.w)));
    }
#pragma unroll
    for (int off = 16; off >= 1; off >>= 1)
        m = fmaxf(m, __shfl_xor(m, off, 32));

    const int wave = tid >> 5;
    const int lane = tid & 31;
    if (lane == 0) red[wave] = m;
    __syncthreads();

    float bm = red[0];
#pragma unroll
    for (int w = 1; w < NWAVES; ++w) bm = fmaxf(bm, red[w]);

    // ---------------- pass 2: exp + weighted sums ----------------------------
    // e = exp2((x - bm) * log2(e));  sx,sy accumulated in raw-coordinate units.
    const float LOG2E = 1.4426950408889634f;
    const float nb    = -bm * LOG2E;

    float s = 0.0f, sx = 0.0f, sy = 0.0f;
#pragma unroll
    for (int i = 0; i < F4_PER_THR; ++i) {
        const int   f4   = i * NTHREADS + tid;
        const float4 v   = t4[f4];
        const int   l    = f4 * 4;                 // linear element index
        const float col0 = (float)(l & (MAP_W - 1));
        const float row  = (float)(l >> 7);

        const float e0 = __builtin_amdgcn_exp2f(fmaf(v.x, LOG2E, nb));
        const float e1 = __builtin_amdgcn_exp2f(fmaf(v.y, LOG2E, nb));
        const float e2 = __builtin_amdgcn_exp2f(fmaf(v.z, LOG2E, nb));
        const float e3 = __builtin_amdgcn_exp2f(fmaf(v.w, LOG2E, nb));

        const float es = (e0 + e1) + (e2 + e3);
        s  += es;
        sy  = fmaf(row,  es, sy);
        sx  = fmaf(col0, es, sx);
        sx += fmaf(2.0f, e2, e1) + 3.0f * e3;      // + (0*e0 + 1*e1 + 2*e2 + 3*e3)
    }

#pragma unroll
    for (int off = 16; off >= 1; off >>= 1) {
        s  += __shfl_xor(s,  off, 32);
        sx += __shfl_xor(sx, off, 32);
        sy += __shfl_xor(sy, off, 32);
    }
    if (lane == 0) {
        red[8  + wave] = s;
        red[16 + wave] = sx;
        red[24 + wave] = sy;
    }
    __syncthreads();

    if (tid == 0) {
        float st = 0.0f, sxt = 0.0f, syt = 0.0f;
#pragma unroll
        for (int w = 0; w < NWAVES; ++w) {
            st  += red[8  + w];
            sxt += red[16 + w];
            syt += red[24 + w];
        }
        const float inv = 1.0f / (127.0f * st);    // fold linspace scale (w/127)
        out[bc * 2 + 0] = sxt * inv;               // expected x (cols)
        out[bc * 2 + 1] = syt * inv;               // expected y (rows)
    }
}

extern "C" void kernel_launch(void* const* d_in, const int* in_sizes, int n_in,
                              void* d_out, int out_size, void* d_ws, size_t ws_size,
                              hipStream_t stream) {
    (void)n_in; (void)d_ws; (void)ws_size; (void)out_size;
    const float* x   = (const float*)d_in[0];
    float*       out = (float*)d_out;

    const int nmaps = in_sizes[0] / MAP_ELEMS;     // B*C = 2048
    const size_t shmem = (size_t)MAP_ELEMS * sizeof(float);  // 64 KB dynamic LDS

    hipLaunchKernelGGL(soft_argmax2d_kernel,
                       dim3(nmaps), dim3(NTHREADS), shmem, stream,
                       x, out, nmaps);
}